// VGAEEncoder_27771258536172
// MI455X (gfx1250) — compile-verified
//
#include <hip/hip_runtime.h>
#include <hip/hip_bf16.h>
#include <stddef.h>

typedef float v2f __attribute__((ext_vector_type(2)));
typedef float v8f __attribute__((ext_vector_type(8)));

#define INDIM 128
#define EMBDIM 64

// ---------------- degree / normalization ----------------

__global__ void deg_init_kernel(float* deg, int n) {
  int i = blockIdx.x * blockDim.x + threadIdx.x;
  if (i < n) deg[i] = 1.0f;  // self-loop weight contributes 1 to every node's degree
}

__global__ void deg_accum_kernel(const int* __restrict__ col, const float* __restrict__ ew,
                                 float* deg, int E) {
  int e = blockIdx.x * blockDim.x + threadIdx.x;
  if (e < E) atomicAdd(&deg[col[e]], ew[e]);
}

__global__ void dinv_kernel(float* deg, int n) {
  int i = blockIdx.x * blockDim.x + threadIdx.x;
  if (i < n) {
    float d = deg[i];
    deg[i] = (d > 0.0f) ? rsqrtf(d) : 0.0f;  // deg becomes dinv in place
  }
}

// out[i][c..c+3] = bias[c..c+3] + dinv[i]^2 * pre[i][c..c+3]  (self-loop + bias, full overwrite)
// C4 = C/4 float4 chunks per row.
__global__ void self_init_kernel(const float4* __restrict__ pre, const float* __restrict__ dinv,
                                 const float4* __restrict__ bias, float4* __restrict__ out,
                                 int n, int C4) {
  long long t = (long long)blockIdx.x * blockDim.x + threadIdx.x;
  long long total = (long long)n * C4;
  if (t >= total) return;
  int i = (int)(t / C4);
  int c = (int)(t % C4);
  float dv = dinv[i];
  float s = dv * dv;
  float4 p = pre[t];
  float4 b = bias[c];
  float4 o;
  o.x = b.x + s * p.x;
  o.y = b.y + s * p.y;
  o.z = b.z + s * p.z;
  o.w = b.w + s * p.w;
  out[t] = o;
}

// ---------------- WMMA GEMM: out = (relu?)(A[n x 128]) @ W[128 x 128] ----------------
// SPLIT: W = [Wa(128x64) | Wb(128x64)], tiles 0-3 -> out0 (ld 64), 4-7 -> out1 (ld 64).
// Each wave computes 16 rows x 128 cols via 8 accumulator tiles, K swept in steps of 4
// with V_WMMA_F32_16X16X4_F32.
// W staged in LDS in fragment-major pair layout: Wp[p][c] = (W[2p][c], W[2p+1][c]),
// so each B fragment is a single contiguous ds_load_b64 (no register repacking).

template <bool RELU, bool SPLIT>
__global__ __launch_bounds__(256) void gemm_wmma_kernel(
    const float* __restrict__ A, const float* __restrict__ Wa, const float* __restrict__ Wb,
    float* __restrict__ out0, float* __restrict__ out1, int nrows) {
  __shared__ v2f Wp[64 * 128];  // 64 KB: [k-pair p][col c] = (W[2p][c], W[2p+1][c])

  for (int idx = threadIdx.x; idx < 64 * 128; idx += 256) {
    int p = idx >> 7;       // k-pair index, rows 2p and 2p+1
    int c = idx & 127;      // column
    float w0, w1;
    if (SPLIT) {
      if (c < 64) {
        w0 = Wa[(2 * p) * 64 + c];
        w1 = Wa[(2 * p + 1) * 64 + c];
      } else {
        w0 = Wb[(2 * p) * 64 + (c - 64)];
        w1 = Wb[(2 * p + 1) * 64 + (c - 64)];
      }
    } else {
      w0 = Wa[(2 * p) * 128 + c];
      w1 = Wa[(2 * p + 1) * 128 + c];
    }
    v2f w;
    w.x = w0;
    w.y = w1;
    Wp[idx] = w;
  }
  __syncthreads();

  const int lane  = threadIdx.x & 31;
  const int wave  = threadIdx.x >> 5;
  const int n     = lane & 15;   // column within 16-wide tile / row M for A frag
  const int khalf = lane >> 4;   // selects K pair {0,1} vs {2,3}
  const int row_base = (blockIdx.x * 8 + wave) * 16;
  if (row_base >= nrows) return;  // nrows is a multiple of 16, tiles are all-or-nothing

  v8f acc[8] = {};

  const float* arow = A + (size_t)(row_base + n) * 128;

  for (int k0 = 0; k0 < 128; k0 += 4) {
    const int kk = k0 + 2 * khalf;       // even
    v2f a;
    if (RELU) {
      a.x = fmaxf(arow[kk], 0.0f);
      a.y = fmaxf(arow[kk + 1], 0.0f);
    } else {
      a = *(const v2f*)(arow + kk);      // 8-byte aligned contiguous pair
    }
    const int pbase = (kk >> 1) * 128 + n;
#pragma unroll
    for (int t = 0; t < 8; ++t) {
      v2f b = Wp[pbase + 16 * t];        // single ds_load_b64
      acc[t] = __builtin_amdgcn_wmma_f32_16x16x4_f32(
          /*neg_a=*/false, a, /*neg_b=*/false, b,
          /*c_mod=*/(short)0, acc[t], /*reuse_a=*/false, /*reuse_b=*/false);
    }
  }

#pragma unroll
  for (int t = 0; t < 8; ++t) {
    float* ob;
    int ldo, cb;
    if (SPLIT) {
      ldo = 64;
      if (t < 4) { ob = out0; cb = 16 * t; }
      else       { ob = out1; cb = 16 * (t - 4); }
    } else {
      ob = out0; ldo = 128; cb = 16 * t;
    }
#pragma unroll
    for (int r = 0; r < 8; ++r) {
      // C/D layout: VGPR r -> M = r (lanes 0-15) or r+8 (lanes 16-31), N = lane%16
      ob[(size_t)(row_base + r + 8 * khalf) * ldo + cb + n] = acc[t][r];
    }
  }
}

// ---------------- edge scatter: warp per edge, float4 per lane ----------------

__global__ void scatter_h_kernel(const int* __restrict__ row, const int* __restrict__ col,
                                 const float* __restrict__ ew, const float* __restrict__ dinv,
                                 const float* __restrict__ hpre, float* __restrict__ out, int E) {
  long long gid = (long long)blockIdx.x * blockDim.x + threadIdx.x;
  int e = (int)(gid >> 5);
  int lane = (int)(gid & 31);
  if (e >= E) return;
  int r = row[e], c = col[e];
  float nrm = dinv[r] * ew[e] * dinv[c];
  float4 v = ((const float4*)(hpre + (size_t)r * 128))[lane];
  float* dst = out + (size_t)c * 128 + lane * 4;
  atomicAdd(dst + 0, nrm * v.x);
  atomicAdd(dst + 1, nrm * v.y);
  atomicAdd(dst + 2, nrm * v.z);
  atomicAdd(dst + 3, nrm * v.w);
}

// lanes 0-15 handle mu (64 ch), lanes 16-31 handle logvar (64 ch)
__global__ void scatter_mulv_kernel(const int* __restrict__ row, const int* __restrict__ col,
                                    const float* __restrict__ ew, const float* __restrict__ dinv,
                                    const float* __restrict__ mupre, const float* __restrict__ lvpre,
                                    float* __restrict__ muout, float* __restrict__ lvout, int E) {
  long long gid = (long long)blockIdx.x * blockDim.x + threadIdx.x;
  int e = (int)(gid >> 5);
  int lane = (int)(gid & 31);
  if (e >= E) return;
  int r = row[e], c = col[e];
  float nrm = dinv[r] * ew[e] * dinv[c];
  const float* pre = (lane < 16) ? mupre : lvpre;
  float* out = (lane < 16) ? muout : lvout;
  int l = lane & 15;
  float4 v = ((const float4*)(pre + (size_t)r * 64))[l];
  float* dst = out + (size_t)c * 64 + l * 4;
  atomicAdd(dst + 0, nrm * v.x);
  atomicAdd(dst + 1, nrm * v.y);
  atomicAdd(dst + 2, nrm * v.z);
  atomicAdd(dst + 3, nrm * v.w);
}

// ---------------- host launch ----------------

extern "C" void kernel_launch(void* const* d_in, const int* in_sizes, int n_in,
                              void* d_out, int out_size, void* d_ws, size_t ws_size,
                              hipStream_t stream) {
  const float* x   = (const float*)d_in[0];
  const int*   ei  = (const int*)d_in[1];   // [2, E] int32 (JAX x64 disabled)
  const float* ew  = (const float*)d_in[2];
  const float* W1  = (const float*)d_in[3];
  const float* b1  = (const float*)d_in[4];
  const float* Wmu = (const float*)d_in[5];
  const float* bmu = (const float*)d_in[6];
  const float* Wlv = (const float*)d_in[7];
  const float* blv = (const float*)d_in[8];

  const int N = in_sizes[0] / INDIM;  // 100000 (multiple of 16)
  const int E = in_sizes[2];          // 1600000
  const int* erow = ei;
  const int* ecol = ei + E;

  // workspace layout (floats): dinv[N] | hpre[N*128] | agg1[N*128]   (~103 MB)
  float* deg   = (float*)d_ws;                 // degree, then dinv in place
  float* hpre  = deg + N;                      // x @ W1 ; later reused for mupre/lvpre
  float* agg1  = hpre + (size_t)N * 128;       // aggregated hidden (pre-relu)
  float* mupre = hpre;                         // relu(agg1) @ Wmu   [N x 64]
  float* lvpre = hpre + (size_t)N * 64;        // relu(agg1) @ Wlv   [N x 64]
  float* muout = (float*)d_out;
  float* lvout = muout + (size_t)N * EMBDIM;

  const int B = 256;

  // 1) symmetric normalization: deg = 1 + sum_in(ew); dinv = rsqrt(deg)
  deg_init_kernel<<<(N + B - 1) / B, B, 0, stream>>>(deg, N);
  deg_accum_kernel<<<(E + B - 1) / B, B, 0, stream>>>(ecol, ew, deg, E);
  dinv_kernel<<<(N + B - 1) / B, B, 0, stream>>>(deg, N);

  // 2) hpre = x @ W1 (fp32 WMMA)
  const int gblocks = (N + 127) / 128;  // 128 rows per 256-thread block (8 waves x 16 rows)
  gemm_wmma_kernel<false, false><<<gblocks, 256, 0, stream>>>(x, W1, nullptr, hpre, nullptr, N);

  // 3) agg1 = b1 + dinv^2*hpre, then scatter norm-weighted edge messages
  long long tot1 = (long long)N * 32;   // float4 chunks (C=128 -> 32)
  self_init_kernel<<<(int)((tot1 + B - 1) / B), B, 0, stream>>>(
      (const float4*)hpre, deg, (const float4*)b1, (float4*)agg1, N, 32);
  long long warps = (long long)E * 32;
  scatter_h_kernel<<<(int)((warps + B - 1) / B), B, 0, stream>>>(erow, ecol, ew, deg, hpre, agg1, E);

  // 4) [mupre|lvpre] = relu(agg1) @ [Wmu|Wlv]  (ReLU fused into A-fragment load)
  gemm_wmma_kernel<true, true><<<gblocks, 256, 0, stream>>>(agg1, Wmu, Wlv, mupre, lvpre, N);

  // 5) outputs: bias + self-loop term, then combined mu/logvar edge scatter
  long long tot2 = (long long)N * 16;   // float4 chunks (C=64 -> 16)
  self_init_kernel<<<(int)((tot2 + B - 1) / B), B, 0, stream>>>(
      (const float4*)mupre, deg, (const float4*)bmu, (float4*)muout, N, 16);
  self_init_kernel<<<(int)((tot2 + B - 1) / B), B, 0, stream>>>(
      (const float4*)lvpre, deg, (const float4*)blv, (float4*)lvout, N, 16);
  scatter_mulv_kernel<<<(int)((warps + B - 1) / B), B, 0, stream>>>(erow, ecol, ew, deg,
                                                                    mupre, lvpre, muout, lvout, E);
}